// AdvancedModel_38354057953685
// MI455X (gfx1250) — compile-verified
//
#include <hip/hip_runtime.h>
#include <hip/hip_bf16.h>

// ---------------------------------------------------------------------------
// CDNA5 (gfx1250) implementation. Key reductions:
//  * trunk layers 8-13 are center-tap 512x512 GEMMs (1x1 spatial)
//  * wf1 uses only tap (3,3); wf2/wf3 are plain GEMMs
//  * bilinear upsample from 1x1 is a broadcast -> sigmoid scalar v[b,c]
//  * Conv2dLocal collapses to windowed sums over wl (bandwidth bound)
//  * Conv3d keeps only kw=10 -> im2col + GEMM
// All GEMM-shaped work goes through a wave32 V_WMMA_F32_16X16X4_F32 kernel
// with a fully branchless inner loop (EXEC stays all-ones, as WMMA requires).
// ---------------------------------------------------------------------------

typedef __attribute__((ext_vector_type(2))) float v2f;
typedef __attribute__((ext_vector_type(8))) float v8f;

// Generic strided fp32 WMMA GEMM: C[m,n] = act( bias[m] + sum_k A[m,k]*B[k,n] )
// One 16x16 tile per wave32 block.  A/B/C addressed via element strides.
__global__ __launch_bounds__(32) void wmma_gemm_f32(
    const float* __restrict__ A, long rsA, long csA,
    const float* __restrict__ B, long rsB, long csB,
    float* __restrict__ C, long rsC, long csC,
    const float* __restrict__ bias, int M, int N, int K, int act)
{
    const int m0   = blockIdx.x * 16;
    const int n0   = blockIdx.y * 16;
    const int lane = threadIdx.x;     // 0..31, full wave
    const int hf   = lane >> 4;       // lane half selects K pair {0,1} vs {2,3}
    const int l    = lane & 15;

    const int  arow = m0 + l;         // A row held by this lane
    const int  bcol = n0 + l;         // B col held by this lane
    const bool aok  = arow < M;
    const bool bok  = bcol < N;
    // clamp OOB lanes to a valid address; value is zero-selected below
    const int  arc  = aok ? arow : (M - 1);
    const int  bcc  = bok ? bcol : (N - 1);

    const float* pA = A + (long)arc * rsA + (long)(hf * 2) * csA;
    const float* pB = B + (long)(hf * 2) * rsB + (long)bcc * csB;
    const long stepA = 4 * csA;
    const long stepB = 4 * rsB;

    // warm the L2 stream for this lane's A row (speculative prefetch)
    __builtin_prefetch(pA, 0, 1);
    __builtin_prefetch(pA + 64 * csA, 0, 1);

    const int Kmain = K & ~3;
    v8f acc = {};
    #pragma unroll 4
    for (int k0 = 0; k0 < Kmain; k0 += 4) {
        float a0 = pA[0], a1 = pA[csA];
        float b0 = pB[0], b1 = pB[rsB];
        v2f a, b;
        a.x = aok ? a0 : 0.f;  a.y = aok ? a1 : 0.f;
        b.x = bok ? b0 : 0.f;  b.y = bok ? b1 : 0.f;
        acc = __builtin_amdgcn_wmma_f32_16x16x4_f32(
            false, a, false, b, (short)0, acc, false, false);
        pA += stepA;
        pB += stepB;
    }
    if (Kmain < K) {              // ragged-K tail: clamp indices, zero-select
        int ka  = Kmain + hf * 2;
        int k0c = (ka     < K) ? ka       : (K - 1);
        int k1c = (ka + 1 < K) ? (ka + 1) : (K - 1);
        float a0 = A[(long)arc * rsA + (long)k0c * csA];
        float a1 = A[(long)arc * rsA + (long)k1c * csA];
        float b0 = B[(long)k0c * rsB + (long)bcc * csB];
        float b1 = B[(long)k1c * rsB + (long)bcc * csB];
        v2f a, b;
        a.x = (aok && ka     < K) ? a0 : 0.f;
        a.y = (aok && ka + 1 < K) ? a1 : 0.f;
        b.x = (bok && ka     < K) ? b0 : 0.f;
        b.y = (bok && ka + 1 < K) ? b1 : 0.f;
        acc = __builtin_amdgcn_wmma_f32_16x16x4_f32(
            false, a, false, b, (short)0, acc, false, false);
    }

    // C/D layout: VGPR r holds rows (r) and (r+8) split across lane halves
    if (bok) {
        #pragma unroll
        for (int r = 0; r < 8; ++r) {
            int row = m0 + r + hf * 8;
            if (row < M) {
                float val = acc[r];
                if (bias) val += bias[row];
                if (act == 1)      val = fmaxf(val, 0.f);
                else if (act == 2) val = 1.f / (1.f + __expf(-val));
                C[(long)row * rsC + (long)bcol * csC] = val;
            }
        }
    }
}

// Direct 3x3 conv, pad = dil, same-size output, fused bias+ReLU (small spatial).
__global__ void conv3x3_relu(const float* __restrict__ in,
                             const float* __restrict__ w,
                             const float* __restrict__ bias,
                             float* __restrict__ out,
                             int Bn, int Cin, int Cout, int H, int W, int dil)
{
    int idx = blockIdx.x * blockDim.x + threadIdx.x;
    int total = Bn * Cout * H * W;
    if (idx >= total) return;
    int x = idx % W; int t = idx / W;
    int y = t % H;   t /= H;
    int oc = t % Cout; int b = t / Cout;

    float acc = bias[oc];
    for (int ci = 0; ci < Cin; ++ci) {
        const float* ip = in + ((long)(b * Cin + ci) * H) * W;
        const float* wp = w + (long)(oc * Cin + ci) * 9;
        #pragma unroll
        for (int ky = 0; ky < 3; ++ky) {
            int yy = y + (ky - 1) * dil;
            if (yy < 0 || yy >= H) continue;
            #pragma unroll
            for (int kx = 0; kx < 3; ++kx) {
                int xx = x + (kx - 1) * dil;
                if (xx < 0 || xx >= W) continue;
                acc += ip[yy * W + xx] * wp[ky * 3 + kx];
            }
        }
    }
    out[idx] = fmaxf(acc, 0.f);
}

__global__ void maxpool2(const float* __restrict__ in, float* __restrict__ out,
                         int Bn, int C, int H, int W)
{
    int Ho = H / 2, Wo = W / 2;
    int idx = blockIdx.x * blockDim.x + threadIdx.x;
    int total = Bn * C * Ho * Wo;
    if (idx >= total) return;
    int x = idx % Wo; int t = idx / Wo;
    int y = t % Ho;   int bc = t / Ho;
    const float* p = in + ((long)bc * H + 2 * y) * W + 2 * x;
    out[idx] = fmaxf(fmaxf(p[0], p[1]), fmaxf(p[W], p[W + 1]));
}

// Locally-connected layer collapsed to windowed sums of wl.
// One block per (i,j,o); lc layout [21, 450] with n = b*225 + i*15 + j.
__global__ __launch_bounds__(256) void local_conv(
    const float* __restrict__ wl, const float* __restrict__ bl,
    const float* __restrict__ v /*[21,2]*/, float* __restrict__ lc)
{
    __shared__ float red[256];
    int blk = blockIdx.x;          // (i*15+j)*21 + o
    int o  = blk % 21;
    int ij = blk / 21;
    int j = ij % 15, i = ij / 15;
    int tid = threadIdx.x;
    int s0 = 25 - i, t0 = 25 - j;  // 14x14 window of taps hitting real pixels

    const float* base = wl + (long)((long)ij * 21 + o) * 21 * 2500;
    float acc0 = 0.f, acc1 = 0.f;
    for (int c = 0; c < 21; ++c) {
        float pv = 0.f;
        if (tid < 196) {
            int s = s0 + tid / 14, t = t0 + tid % 14;
            pv = base[(long)c * 2500 + s * 50 + t];
        }
        red[tid] = pv;
        __syncthreads();
        for (int st = 128; st > 0; st >>= 1) {
            if (tid < st) red[tid] += red[tid + st];
            __syncthreads();
        }
        float S = red[0];
        acc0 += v[c * 2 + 0] * S;
        acc1 += v[c * 2 + 1] * S;
        __syncthreads();
    }
    if (tid == 0) {
        float bb = bl[o * 225 + ij];
        lc[(long)o * 450 +       ij] = acc0 + bb;
        lc[(long)o * 450 + 225 + ij] = acc1 + bb;
    }
}

// im2col for the effective 9x9 conv (Conv3d with only kw=10 live).
// h layout [21, 450]; col layout [1701, 450].
__global__ void im2col_9x9(const float* __restrict__ h, float* __restrict__ col)
{
    int idx = blockIdx.x * blockDim.x + threadIdx.x;
    const int total = 1701 * 450;
    if (idx >= total) return;
    int n = idx % 450; int k = idx / 450;
    int b = n / 225;   int p = n % 225;
    int d = p / 15,    hh = p % 15;
    int c = k / 81;    int r = k % 81;
    int kd = r / 9,    kh = r % 9;
    int dd = d + kd - 4, h2 = hh + kh - 4;
    float vv = 0.f;
    if (dd >= 0 && dd < 15 && h2 >= 0 && h2 < 15)
        vv = h[(long)c * 450 + b * 225 + dd * 15 + h2];
    col[idx] = vv;
}

// Block-min over W (groups of 5) then softmax over 105 channels.
// in layout [105, 450]; out layout [2, 105, 15, 3].
__global__ __launch_bounds__(128) void minpool_softmax(
    const float* __restrict__ in, float* __restrict__ out)
{
    __shared__ float red[128];
    int blk = blockIdx.x;            // b*45 + i*3 + jo
    int jo = blk % 3; int t = blk / 3;
    int i = t % 15;   int b = t / 15;
    int tid = threadIdx.x;
    bool active = tid < 105;

    float myv = -3.0e38f;
    if (active) {
        const float* p = in + (long)tid * 450 + b * 225 + i * 15 + jo * 5;
        float mn = p[0];
        mn = fminf(mn, p[1]); mn = fminf(mn, p[2]);
        mn = fminf(mn, p[3]); mn = fminf(mn, p[4]);
        myv = mn;
    }
    red[tid] = active ? myv : -3.0e38f;
    __syncthreads();
    for (int st = 64; st > 0; st >>= 1) {
        if (tid < st) red[tid] = fmaxf(red[tid], red[tid + st]);
        __syncthreads();
    }
    float mx = red[0];
    __syncthreads();
    float e = active ? __expf(myv - mx) : 0.f;
    red[tid] = e;
    __syncthreads();
    for (int st = 64; st > 0; st >>= 1) {
        if (tid < st) red[tid] += red[tid + st];
        __syncthreads();
    }
    float s = red[0];
    if (active)
        out[((long)(b * 105 + tid) * 15 + i) * 3 + jo] = e / s;
}

// ---------------------------------------------------------------------------

static inline void launch_gemm(hipStream_t s,
    const float* A, long rsA, long csA,
    const float* B, long rsB, long csB,
    float* C, long rsC, long csC,
    const float* bias, int M, int N, int K, int act)
{
    dim3 g((M + 15) / 16, (N + 15) / 16);
    wmma_gemm_f32<<<g, 32, 0, s>>>(A, rsA, csA, B, rsB, csB, C, rsC, csC,
                                   bias, M, N, K, act);
}

extern "C" void kernel_launch(void* const* d_in, const int* in_sizes, int n_in,
                              void* d_out, int out_size, void* d_ws, size_t ws_size,
                              hipStream_t stream)
{
    const float* x = (const float*)d_in[0];
    const float* w[13]; const float* bv[13];
    for (int i = 0; i < 13; ++i) {
        w[i]  = (const float*)d_in[1 + 2 * i];
        bv[i] = (const float*)d_in[2 + 2 * i];
    }
    const float* wf1 = (const float*)d_in[27]; const float* bf1 = (const float*)d_in[28];
    const float* wf2 = (const float*)d_in[29]; const float* bf2 = (const float*)d_in[30];
    const float* wf3 = (const float*)d_in[31]; const float* bf3 = (const float*)d_in[32];
    const float* wl  = (const float*)d_in[33]; const float* bl  = (const float*)d_in[34];
    const float* wa  = (const float*)d_in[35]; const float* ba  = (const float*)d_in[36];
    const float* wm  = (const float*)d_in[37]; const float* bm  = (const float*)d_in[38];
    const float* wb  = (const float*)d_in[39]; const float* bb  = (const float*)d_in[40];

    float* ws  = (float*)d_ws;
    float* tA  = ws;              // ping  (<= 2*64*14*14 = 25088)
    float* tB  = ws + 32768;      // pong
    float* fc1 = ws + 65536;      // [4096,2]
    float* fc2 = ws + 73728;      // [4096,2]
    float* v   = ws + 81920;      // [21,2] sigmoid scalars
    float* lc  = ws + 82048;      // [21,450] local-conv out
    float* wao = ws + 91648;      // [21,450] after wa
    float* col = ws + 101376;     // [1701,450] im2col
    float* cd  = ws + 866880;     // [105,450] conv3d out
    float* wbo = ws + 914176;     // [105,450] after wb

    auto cdiv = [](int a, int b) { return (a + b - 1) / b; };

    // ---- VGG trunk (direct convs at 14/7/3 spatial) ----
    int tot;
    tot = 2 * 64 * 14 * 14;
    conv3x3_relu<<<cdiv(tot, 256), 256, 0, stream>>>(x,  w[0], bv[0], tA, 2,   3,  64, 14, 14, 1);
    conv3x3_relu<<<cdiv(tot, 256), 256, 0, stream>>>(tA, w[1], bv[1], tB, 2,  64,  64, 14, 14, 1);
    tot = 2 * 64 * 7 * 7;
    maxpool2<<<cdiv(tot, 256), 256, 0, stream>>>(tB, tA, 2, 64, 14, 14);          // -> 7x7
    tot = 2 * 128 * 7 * 7;
    conv3x3_relu<<<cdiv(tot, 256), 256, 0, stream>>>(tA, w[2], bv[2], tB, 2,  64, 128, 7, 7, 1);
    conv3x3_relu<<<cdiv(tot, 256), 256, 0, stream>>>(tB, w[3], bv[3], tA, 2, 128, 128, 7, 7, 1);
    tot = 2 * 128 * 3 * 3;
    maxpool2<<<cdiv(tot, 256), 256, 0, stream>>>(tA, tB, 2, 128, 7, 7);           // -> 3x3
    tot = 2 * 256 * 3 * 3;
    conv3x3_relu<<<cdiv(tot, 256), 256, 0, stream>>>(tB, w[4], bv[4], tA, 2, 128, 256, 3, 3, 1);
    conv3x3_relu<<<cdiv(tot, 256), 256, 0, stream>>>(tA, w[5], bv[5], tB, 2, 256, 256, 3, 3, 1);
    conv3x3_relu<<<cdiv(tot, 256), 256, 0, stream>>>(tB, w[6], bv[6], tA, 2, 256, 256, 3, 3, 1);
    tot = 2 * 256;
    maxpool2<<<cdiv(tot, 256), 256, 0, stream>>>(tA, tB, 2, 256, 3, 3);           // -> 1x1 [2,256]

    // ---- layers 8..13 at 1x1 spatial: center-tap GEMMs (WMMA) ----
    // A[m,k] = w[(m*Cin+k)*9 + 4]; B[k,n] = act[n*Cin+k]; C[m,n] = out[n*512+m]
    launch_gemm(stream, w[7]  + 4, 256 * 9, 9, tB, 1, 256, tA, 1, 512, bv[7],  512, 2, 256, 1);
    launch_gemm(stream, w[8]  + 4, 512 * 9, 9, tA, 1, 512, tB, 1, 512, bv[8],  512, 2, 512, 1);
    launch_gemm(stream, w[9]  + 4, 512 * 9, 9, tB, 1, 512, tA, 1, 512, bv[9],  512, 2, 512, 1);
    launch_gemm(stream, w[10] + 4, 512 * 9, 9, tA, 1, 512, tB, 1, 512, bv[10], 512, 2, 512, 1);
    launch_gemm(stream, w[11] + 4, 512 * 9, 9, tB, 1, 512, tA, 1, 512, bv[11], 512, 2, 512, 1);
    launch_gemm(stream, w[12] + 4, 512 * 9, 9, tA, 1, 512, tB, 1, 512, bv[12], 512, 2, 512, 1);

    // ---- FC head: wf1 tap(3,3) mat-vec, wf2, wf3 (+sigmoid == upsample+sigmoid) ----
    launch_gemm(stream, wf1 + 24, 512L * 49, 49, tB, 1, 512, fc1, 2, 1, bf1, 4096, 2, 512,  1);
    launch_gemm(stream, wf2,      4096,       1, fc1, 2, 1,  fc2, 2, 1, bf2, 4096, 2, 4096, 1);
    launch_gemm(stream, wf3,      4096,       1, fc2, 2, 1,  v,   2, 1, bf3,   21, 2, 4096, 2);

    // ---- locally-connected layer: windowed sums of wl (bandwidth-dominant) ----
    local_conv<<<225 * 21, 256, 0, stream>>>(wl, bl, v, lc);

    // ---- wa linear over labels ----
    launch_gemm(stream, wa, 21, 1, lc, 450, 1, wao, 450, 1, ba, 21, 450, 21, 0);

    // ---- Conv3d head (kw=10 only) as im2col + GEMM ----
    im2col_9x9<<<cdiv(1701 * 450, 256), 256, 0, stream>>>(wao, col);
    launch_gemm(stream, wm + 10, 1701L * 21, 21, col, 450, 1, cd, 450, 1, bm, 105, 450, 1701, 0);

    // ---- wb linear ----
    launch_gemm(stream, wb, 105, 1, cd, 450, 1, wbo, 450, 1, bb, 105, 450, 105, 0);

    // ---- block-min over W (M=5) + softmax over channels ----
    minpool_softmax<<<2 * 15 * 3, 128, 0, stream>>>(wbo, (float*)d_out);
}